// FM_80719615361479
// MI455X (gfx1250) — compile-verified
//
#include <hip/hip_runtime.h>
#include <math.h>

typedef __attribute__((ext_vector_type(2))) float v2f;
typedef __attribute__((ext_vector_type(8))) float v8f;

#define B_SZ 1024
#define F_SZ 512
#define K_SZ 16
#define WAVES_PER_BLOCK 8

// One wave32 per batch row b.
//   p2[b] = trace( C(16x512) * A(512x16) )   via 128x V_WMMA_F32_16X16X4_F32
//   A[r,k] = x[r]*W[r,k]
//   C[k,16g+m] = x[32k+g]*Wpre[32k+g,m] + G_k[g]   (G_k = running exclusive prefix, kept in a register)
__global__ __launch_bounds__(256) void FM_80719615361479_kernel(
    const float* __restrict__ x, const float* __restrict__ W,
    const float* __restrict__ lin_w, const float* __restrict__ lin_b,
    float* __restrict__ out)
{
    __shared__ float sWpre[F_SZ][17];              // inclusive prefix of W along k, padded row
    __shared__ float sX[WAVES_PER_BLOCK][F_SZ];    // one x row per wave

    const int tid  = threadIdx.x;
    const int lane = tid & 31;
    const int w    = tid >> 5;            // wave id in block
    const int ln   = lane & 15;
    const int hi   = lane >> 4;           // half-wave select (K=0,1 vs K=2,3 of WMMA operands)
    const int b    = blockIdx.x * WAVES_PER_BLOCK + w;

    // ---- stage W prefix sums (block-cooperative; W is L2-resident, 32KB) ----
    for (int f = tid; f < F_SZ; f += 256) {
        const float* wrow = W + f * K_SZ;
        float run = 0.f;
        #pragma unroll
        for (int m = 0; m < K_SZ; ++m) { run += wrow[m]; sWpre[f][m] = run; }
    }

    // ---- stage this wave's x row, fold in the linear term partial ----
    const float* xrow = x + (size_t)b * F_SZ;
    float p1 = 0.f;
    #pragma unroll
    for (int i = 0; i < 16; ++i) {
        float v = xrow[i * 32 + lane];
        sX[w][i * 32 + lane] = v;
        p1 = fmaf(v, lin_w[i * 32 + lane], p1);
    }
    __syncthreads();

    // ---- D = C(16x512) x A(512x16), K-chunks of 4, all-f32 WMMA ----
    // A-operand 16x4 layout: lanes 0-15 hold K=0(v0),K=1(v1); lanes 16-31 hold K=2,K=3. M = ln.
    // B-operand 4x16 assumed mirror: lanes 0-15 rows K=0(v0),K=1(v1); lanes 16-31 rows K=2,K=3. N = ln.
    v8f acc = {};
    float G = 0.f;                 // exclusive prefix G_k[g] for k = ln
    const int k = ln;
    for (int g = 0; g < 32; ++g) {
        const int   fa    = 32 * k + g;
        const float xk    = sX[w][fa];
        const float* wprow = sWpre[fa];
        #pragma unroll
        for (int jj = 0; jj < 4; ++jj) {
            const int r0 = g * 16 + jj * 4;       // inner-K base (r index into 0..511)
            const int ra = r0 + 2 * hi;           // this lane's two r's
            const int rb = ra + 1;
            const int c0 = jj * 4 + 2 * hi;       // = ra & 15

            // C[k, ra], C[k, rb]
            float a0 = fmaf(xk, wprow[c0],     G);
            float a1 = fmaf(xk, wprow[c0 + 1], G);

            // A[ra, ln], A[rb, ln]  with W recovered by differencing Wpre (branch-free)
            float wpa  = sWpre[ra][ln];
            float wpam = sWpre[ra][(ln + 15) & 15];
            float wpb  = sWpre[rb][ln];
            float wpbm = sWpre[rb][(ln + 15) & 15];
            float b0 = sX[w][ra] * (wpa - (ln ? wpam : 0.f));
            float b1 = sX[w][rb] * (wpb - (ln ? wpbm : 0.f));

            v2f Am = {a0, a1};
            v2f Bm = {b0, b1};
            acc = __builtin_amdgcn_wmma_f32_16x16x4_f32(
                false, Am, false, Bm, (short)0, acc, false, false);
        }
        G = fmaf(xk, wprow[15], G);   // fold rowsum(W[fa]) -> exclusive prefix for g+1
    }

    // ---- trace: lane holds D[i + 8*hi, ln] in acc[i]; diagonal where i + 8*hi == ln ----
    float p2p = 0.f;
    #pragma unroll
    for (int i = 0; i < 8; ++i)
        if (ln == i + 8 * hi) p2p = acc[i];

    // ---- wave32 reductions, fuse sigmoid ----
    #pragma unroll
    for (int off = 16; off > 0; off >>= 1) {
        p2p += __shfl_xor(p2p, off, 32);
        p1  += __shfl_xor(p1,  off, 32);
    }
    if (lane == 0) {
        float z = p1 + lin_b[0] + p2p;
        out[b] = 1.0f / (1.0f + expf(-z));
    }
}

extern "C" void kernel_launch(void* const* d_in, const int* in_sizes, int n_in,
                              void* d_out, int out_size, void* d_ws, size_t ws_size,
                              hipStream_t stream) {
    (void)in_sizes; (void)n_in; (void)d_ws; (void)ws_size; (void)out_size;
    const float* x   = (const float*)d_in[0];
    const float* W   = (const float*)d_in[1];
    const float* lw  = (const float*)d_in[2];
    const float* lb  = (const float*)d_in[3];
    float* out = (float*)d_out;

    dim3 grid(B_SZ / WAVES_PER_BLOCK);
    dim3 block(32 * WAVES_PER_BLOCK);
    FM_80719615361479_kernel<<<grid, block, 0, stream>>>(x, W, lw, lb, out);
}